// Mapping_3968549781917
// MI455X (gfx1250) — compile-verified
//
#include <hip/hip_runtime.h>
#include <hip/hip_bf16.h>
#include <math.h>

// ---------------------------------------------------------------------------
// MI455X (gfx1250) semantic-mapping pipeline.
// Memory-bound workload: scatter targets (~34 MB) kept L2-resident (192 MB L2),
// z-range reductions mapped onto V_WMMA_F32_16X16X4_F32 (masked-ones matmul),
// rotate+translate warps fused exactly (bilinear-of-bilinear at grid points).
// ---------------------------------------------------------------------------

typedef __attribute__((ext_vector_type(2))) float v2f;
typedef __attribute__((ext_vector_type(8))) float v8f;

#define BSZ      8
#define CALL     20
#define NSEM     16
#define FH       480
#define FW       640
#define MAPM     480
#define VRD      100      // voxel x/y dim
#define NZ       88       // voxel z dim
#define MINZ     23       // int(35/5 - (-16))
#define MAXZ     33       // int(89/5 - (-16))
#define FLOORZ   9        // int(-35/5 - (-16))
#define MVSTD    28       // int(MIN_VISION // 5)
#define XWIN     190      // x1 = 240 - 50
#define YWIN     240      // y1
#define DEGC     57.29577951308232f
#define PIF      3.14159265358979323846f

// workspace layout (floats)
#define OFF_PARAMS 0                      // 8 batches * 8 floats
#define OFF_V0     64                     // 8 * 10000 cells * 88 z
#define OFF_SEM    (64 + 7040000)         // 8 * 16 * 10000
#define OFF_FPMAP  (OFF_SEM + 1280000)    // 8 * 10000
#define OFF_FPEXP  (OFF_FPMAP + 80000)    // 8 * 10000
#define ZERO_COUNT 8320000                // V0 + sem region
#define POSE_OFF   36944000               // 80000 + 8*20*480*480

__device__ __forceinline__ void atomAdd(float* p, float v) {
  __hip_atomic_fetch_add(p, v, __ATOMIC_RELAXED, __HIP_MEMORY_SCOPE_AGENT);
}

// --------------------------------------------------------------------------
__global__ void zero_kernel(float4* p, int n4) {
  int i = blockIdx.x * blockDim.x + threadIdx.x;
  if (i < n4) p[i] = make_float4(0.f, 0.f, 0.f, 0.f);
}

// --------------------------------------------------------------------------
// Pose composition + depth-row mask, one block per batch.
__global__ void pose_kernel(const float* __restrict__ obs,
                            const float* __restrict__ pose_obs,
                            const float* __restrict__ poses_last,
                            float* __restrict__ wsf,
                            float* __restrict__ out) {
  int b = blockIdx.x;
  int tid = threadIdx.x;
  __shared__ int cnt;
  if (tid == 0) cnt = 0;
  __syncthreads();

  const float FOCAL = 320.f / tanf(0.6894050545f);   // 39.5 deg
  const float MINV  = 88.f * FOCAL / 240.f;          // 88 / tan(atan(240/F))
  int local = 0;
  for (int col = tid; col < FW; col += blockDim.x) {
    float d  = obs[(((size_t)b * CALL + 3) * FH + (FH - 1)) * FW + col];
    float re = (d < 3000.f) ? d : MINV;
    if (re - MINV - 60.f > 0.f) local++;
  }
  atomicAdd(&cnt, local);
  __syncthreads();

  if (tid == 0) {
    float th  = poses_last[b * 3 + 2] / DEGC;
    float p0  = pose_obs[b * 3 + 0];
    float p1  = pose_obs[b * 3 + 1];
    float p2  = pose_obs[b * 3 + 2];
    float ny  = poses_last[b * 3 + 1] + p0 * sinf(th) + p1 * cosf(th);
    float nx  = poses_last[b * 3 + 0] + p0 * cosf(th) - p1 * sinf(th);
    float nt  = poses_last[b * 3 + 2] + p2 * DEGC;
    nt = fmodf(nt - 180.f, 360.f) + 180.f;
    nt = fmodf(nt + 180.f, 360.f) - 180.f;

    out[POSE_OFF + b * 3 + 0] = nx;
    out[POSE_OFF + b * 3 + 1] = ny;
    out[POSE_OFF + b * 3 + 2] = nt;
    out[POSE_OFF + 24 + b * 3 + 0] = nx;
    out[POSE_OFF + 24 + b * 3 + 1] = ny;
    out[POSE_OFF + 24 + b * 3 + 2] = nt;

    float stx = -(nx * 20.f - 240.f) / 240.f;
    float sty = -(ny * 20.f - 240.f) / 240.f;
    float t   = (90.f - nt) * PIF / 180.f;
    float* prm = wsf + OFF_PARAMS + b * 8;
    prm[0] = cosf(t);
    prm[1] = sinf(t);
    prm[2] = stx;
    prm[3] = sty;
    prm[4] = (cnt > 160) ? 1.f : 0.f;   // count > W/4
  }
}

// --------------------------------------------------------------------------
// Trilinear splat. Channel 0 -> full z-resolved grid (for WMMA reductions),
// sem channels -> directly into the [MINZ,MAXZ) projection (all consumers
// only ever use that range-sum).  All targets are L2-resident.
__global__ void splat_kernel(const float* __restrict__ obs,
                             const float* __restrict__ agent_heights,
                             float* __restrict__ wsf) {
  int idx = blockIdx.x * blockDim.x + threadIdx.x;
  int b   = idx / (FH * FW);
  int p   = idx % (FH * FW);
  int r   = p / FW;
  int cc  = p % FW;

  const float FOCAL = 320.f / tanf(0.6894050545f);
  float Y  = obs[(((size_t)b * CALL + 3) * FH + r) * FW + cc];     // depth
  float gz = (float)(FH - 1 - r);
  float X  = ((float)cc - 319.5f) * Y / FOCAL + 250.f;             // + SHIFT_X
  float Z  = (gz - 239.5f) * Y / FOCAL + agent_heights[b] * 100.f;

  float posx = X * 0.2f;          // X / RES
  float posy = Y * 0.2f;
  float posz = Z * 0.2f + 16.f;   // Z / Z_RES - MIN_H

  int x0 = (int)floorf(posx), y0 = (int)floorf(posy), z0 = (int)floorf(posz);
  float wx[2], wy[2], wz[2];
  int   px[2], py[2], pz[2];
#pragma unroll
  for (int k = 0; k < 2; ++k) {
    int qx = x0 + k; bool sx = (qx > 0) && (qx < VRD);
    wx[k] = sx ? (1.f - fabsf(posx - (float)qx)) : 0.f; px[k] = sx ? qx : 0;
    int qy = y0 + k; bool sy = (qy > 0) && (qy < VRD);
    wy[k] = sy ? (1.f - fabsf(posy - (float)qy)) : 0.f; py[k] = sy ? qy : 0;
    int qz = z0 + k; bool sz = (qz > 0) && (qz < NZ);
    wz[k] = sz ? (1.f - fabsf(posz - (float)qz)) : 0.f; pz[k] = sz ? qz : 0;
  }

  float* V0 = wsf + OFF_V0 + (size_t)b * (VRD * VRD * NZ);
  // channel 0 (feat == 1): full z-resolved grid, cell = py*100 + px (swapaxes)
#pragma unroll
  for (int iy = 0; iy < 2; ++iy)
#pragma unroll
    for (int ix = 0; ix < 2; ++ix) {
      float wxy = wx[ix] * wy[iy];
      if (wxy == 0.f) continue;
      int cell = py[iy] * VRD + px[ix];
#pragma unroll
      for (int iz = 0; iz < 2; ++iz) {
        float w = wxy * wz[iz];
        if (w != 0.f) atomAdd(V0 + (size_t)cell * NZ + pz[iz], w);
      }
    }

  // sem channels: only the [MINZ,MAXZ) z-range survives downstream
  float wzin = 0.f;
#pragma unroll
  for (int iz = 0; iz < 2; ++iz)
    if (pz[iz] >= MINZ && pz[iz] < MAXZ) wzin += wz[iz];
  if (wzin == 0.f) return;

  float sv[NSEM];
#pragma unroll
  for (int s = 0; s < NSEM; ++s)
    sv[s] = obs[(((size_t)b * CALL + 4 + s) * FH + r) * FW + cc];

  float* SP = wsf + OFF_SEM + (size_t)b * (NSEM * VRD * VRD);
#pragma unroll
  for (int iy = 0; iy < 2; ++iy)
#pragma unroll
    for (int ix = 0; ix < 2; ++ix) {
      float wxy = wx[ix] * wy[iy] * wzin;
      if (wxy == 0.f) continue;
      int cell = py[iy] * VRD + px[ix];
#pragma unroll
      for (int s = 0; s < NSEM; ++s)
        atomAdd(SP + s * (VRD * VRD) + cell, wxy * sv[s]);
    }
}

// --------------------------------------------------------------------------
// Four z-range reductions as one WMMA accumulate chain:
// A (16x4 f32) = 16 voxel columns x 4 z-slices, B (4x16) columns 0..3 are the
// 0/1 range masks {agent, all, around-floor, under-floor}. 22 chained
// V_WMMA_F32_16X16X4_F32 steps compute all four range sums for 16 cells.
//
// B-construction is fully branchless: over z the column values take only four
// distinct per-lane combinations (z ranges [0,9) [9,23) [23,33) [33,88)),
// each a sum of the lane's column-selector bits.  These four floats are
// precomputed once per lane; the loop body selects among them with compares
// against constants -> pure v_cndmask, no EXEC manipulation between WMMAs.
#if defined(__has_builtin)
#if __has_builtin(__builtin_amdgcn_wmma_f32_16x16x4_f32)
#define HAVE_WMMA_F32X4 1
#endif
#endif

__device__ __forceinline__ float bsel(int z, float bLow, float bFloor,
                                      float bAgent, float bHigh) {
  float r = (z < FLOORZ) ? bLow : bFloor;   // [0,9) vs [9,23)
  r = (z >= MINZ) ? bAgent : r;             // [23,33)
  r = (z >= MAXZ) ? bHigh : r;              // [33,88)
  return r;
}

__device__ __forceinline__ void emit_cell(float* __restrict__ wsf,
                                          float* __restrict__ out,
                                          int b, int cell, float maskv,
                                          float agentS, float allS,
                                          float aroundS, float floorS) {
  float under = (floorS == 0.f) ? aroundS : 0.f;
  int pyc = cell / VRD, pxc = cell - pyc * VRD;
  if (pyc == MVSTD && pxc >= 47 && pxc < 53)
    under = (maskv > 0.5f) ? 1.f : under;
  float fpm = fminf(fmaxf(agentS + under, 0.f), 1.f);   // MAP_THR = 1
  float fpe = fminf(fmaxf(allS, 0.f), 1.f);             // EXP_THR = 1
  wsf[OFF_FPMAP + b * (VRD * VRD) + cell] = fpm;
  wsf[OFF_FPEXP + b * (VRD * VRD) + cell] = fpe;
  out[b * (VRD * VRD) + cell] = fpm;                    // fp_map_pred output
}

__global__ void proj_wmma_kernel(float* __restrict__ wsf,
                                 float* __restrict__ out) {
  int lane = threadIdx.x & 31;
  int wave = blockIdx.x * (blockDim.x >> 5) + (threadIdx.x >> 5);
  int b        = wave / 625;             // 625 waves of 16 cells per batch
  int cellBase = (wave % 625) * 16;
  float maskv  = wsf[OFF_PARAMS + b * 8 + 4];
  const float* V0 = wsf + OFF_V0 + (size_t)b * (VRD * VRD * NZ);

#if defined(HAVE_WMMA_F32X4)
  int m    = lane & 15;                  // A: row M / B: column N
  int koff = (lane >> 4) << 1;           // lanes 16..31 carry K = 2,3
  const float* vb = V0 + (size_t)(cellBase + m) * NZ;

  // per-lane column-selector bits and the four z-range B values
  float s0 = (m == 0) ? 1.f : 0.f;       // agent-height column
  float s1 = (m == 1) ? 1.f : 0.f;       // all-height column
  float s2 = (m == 2) ? 1.f : 0.f;       // around-floor column
  float s3 = (m == 3) ? 1.f : 0.f;       // under-floor column
  float bLow   = s1 + s2;                // z in [0,9)
  float bFloor = s1 + s2 + s3;           // z in [9,23)
  float bAgent = s0 + s1;                // z in [23,33)
  float bHigh  = s1;                     // z in [33,88)

  v8f acc = {};
#pragma unroll
  for (int k0 = 0; k0 < NZ; k0 += 4) {
    int ka = k0 + koff;
    v2f A = *(const v2f*)(vb + ka);
    A.x = rintf(A.x);                    // reference rounds voxel contents
    A.y = rintf(A.y);
    v2f B;
    B.x = bsel(ka,     bLow, bFloor, bAgent, bHigh);
    B.y = bsel(ka + 1, bLow, bFloor, bAgent, bHigh);
    acc = __builtin_amdgcn_wmma_f32_16x16x4_f32(
        false, A, false, B, (short)0, acc, false, false);
  }
  // D layout: vgpr r, lanes 0-15 -> (M=r, N=lane); lanes 16-31 -> (M=8+r).
  int base = lane & 16;
#pragma unroll
  for (int r = 0; r < 8; ++r) {
    float v  = acc[r];
    float c0 = __shfl(v, base + 0, 32);
    float c1 = __shfl(v, base + 1, 32);
    float c2 = __shfl(v, base + 2, 32);
    float c3 = __shfl(v, base + 3, 32);
    if (m == 0) {
      int cell = cellBase + r + ((lane >> 4) ? 8 : 0);
      emit_cell(wsf, out, b, cell, maskv, c0, c1, c2, c3);
    }
  }
#else
  // VALU fallback: lanes 0..15 each reduce one cell serially.
  if (lane < 16) {
    const float* vb = V0 + (size_t)(cellBase + lane) * NZ;
    float t0 = 0.f, t1 = 0.f, t2 = 0.f, t3 = 0.f;
    for (int z = 0; z < NZ; ++z) {
      float v = rintf(vb[z]);
      t0 += (z >= MINZ && z < MAXZ) ? v : 0.f;
      t1 += v;
      t2 += (z < MINZ) ? v : 0.f;
      t3 += (z >= FLOORZ && z < MINZ) ? v : 0.f;
    }
    emit_cell(wsf, out, b, cellBase + lane, maskv, t0, t1, t2, t3);
  }
#endif
}

// --------------------------------------------------------------------------
// Fused rotate+translate+max. Exact: the translate warp samples the rotate
// warp's output at integer grid points, so each output pixel is a 4x4-tap
// bilinear-of-bilinear evaluation of the virtual agent_view (zero outside the
// 100x100 window). ~96% of pixels early-out to a pure maps_last copy.
__global__ void warp_fuse_kernel(const float* __restrict__ maps,
                                 const float* __restrict__ wsf,
                                 float* __restrict__ out) {
  int idx = blockIdx.x * blockDim.x + threadIdx.x;
  int b   = idx / (MAPM * MAPM);
  int pix = idx % (MAPM * MAPM);
  int i   = pix / MAPM;
  int j   = pix % MAPM;

  __builtin_prefetch(maps + (size_t)b * CALL * (MAPM * MAPM) + pix, 0, 3);

  const float* prm = wsf + OFF_PARAMS + b * 8;
  float cosT = prm[0], sinT = prm[1], stx = prm[2], sty = prm[3];
  const float* fpM = wsf + OFF_FPMAP + b * (VRD * VRD);
  const float* fpE = wsf + OFF_FPEXP + b * (VRD * VRD);
  const float* SP  = wsf + OFF_SEM + (size_t)b * (NSEM * VRD * VRD);

  float acc[CALL];
#pragma unroll
  for (int c = 0; c < CALL; ++c) acc[c] = 0.f;

  float xn = (2.f * (float)j + 1.f) / (float)MAPM - 1.f;
  float yn = (2.f * (float)i + 1.f) / (float)MAPM - 1.f;
  float x  = (xn + stx + 1.f) * 0.5f * (float)(MAPM - 1);
  float y  = (yn + sty + 1.f) * 0.5f * (float)(MAPM - 1);
  int   x0 = (int)floorf(x), y0 = (int)floorf(y);
  float fx = x - (float)x0,  fy = y - (float)y0;

#pragma unroll
  for (int cy = 0; cy < 2; ++cy)
#pragma unroll
    for (int cx = 0; cx < 2; ++cx) {
      int ix2 = x0 + cx, iy2 = y0 + cy;
      float w2 = (cx ? fx : 1.f - fx) * (cy ? fy : 1.f - fy);
      if (w2 == 0.f || ix2 < 0 || ix2 >= MAPM || iy2 < 0 || iy2 >= MAPM)
        continue;
      // virtual rotated(iy2, ix2): bilinear sample of agent_view at R * n
      float xn2 = (2.f * (float)ix2 + 1.f) / (float)MAPM - 1.f;
      float yn2 = (2.f * (float)iy2 + 1.f) / (float)MAPM - 1.f;
      float gx  = cosT * xn2 - sinT * yn2;
      float gy  = sinT * xn2 + cosT * yn2;
      float xs  = (gx + 1.f) * 0.5f * (float)(MAPM - 1);
      float ys  = (gy + 1.f) * 0.5f * (float)(MAPM - 1);
      int   xs0 = (int)floorf(xs), ys0 = (int)floorf(ys);
      float gfx = xs - (float)xs0, gfy = ys - (float)ys0;
#pragma unroll
      for (int dy = 0; dy < 2; ++dy)
#pragma unroll
        for (int dx = 0; dx < 2; ++dx) {
          int ixs = xs0 + dx, iys = ys0 + dy;
          float w1 = (dx ? gfx : 1.f - gfx) * (dy ? gfy : 1.f - gfy);
          if (w1 == 0.f || ixs < 0 || ixs >= MAPM || iys < 0 || iys >= MAPM)
            continue;
          int vy = iys - YWIN, vx = ixs - XWIN;
          if (vy < 0 || vy >= VRD || vx < 0 || vx >= VRD) continue;
          int   cell = vy * VRD + vx;
          float w    = w2 * w1;
          acc[0] += w * fpM[cell];
          acc[1] += w * fpE[cell];
#pragma unroll
          for (int s = 0; s < NSEM; ++s)
            acc[4 + s] += w * fminf(fmaxf(SP[s * (VRD * VRD) + cell] * 0.2f,
                                          0.f), 1.f);   // CAT_THR = 5
        }
    }

#pragma unroll
  for (int c = 0; c < CALL; ++c) {
    size_t mi = ((size_t)b * CALL + c) * (MAPM * MAPM) + pix;
    out[80000 + mi] = fmaxf(maps[mi], acc[c]);
  }
}

// --------------------------------------------------------------------------
extern "C" void kernel_launch(void* const* d_in, const int* in_sizes, int n_in,
                              void* d_out, int out_size, void* d_ws,
                              size_t ws_size, hipStream_t stream) {
  const float* obs           = (const float*)d_in[0];
  const float* pose_obs      = (const float*)d_in[1];
  const float* maps_last     = (const float*)d_in[2];
  const float* poses_last    = (const float*)d_in[3];
  const float* agent_heights = (const float*)d_in[4];
  float* out = (float*)d_out;
  float* wsf = (float*)d_ws;

  // 1) zero the accumulation buffers (V0 grid + sem projections)
  zero_kernel<<<ZERO_COUNT / 4 / 256, 256, 0, stream>>>(
      (float4*)(wsf + OFF_V0), ZERO_COUNT / 4);
  // 2) pose composition, warp params, depth-row mask
  pose_kernel<<<BSZ, 64, 0, stream>>>(obs, pose_obs, poses_last, wsf, out);
  // 3) trilinear splat (L2-resident atomic adds)
  splat_kernel<<<BSZ * FH * FW / 256, 256, 0, stream>>>(obs, agent_heights,
                                                        wsf);
  // 4) z-range reductions on the matrix pipe (5000 waves, 16 cells each)
  proj_wmma_kernel<<<625, 256, 0, stream>>>(wsf, out);
  // 5) fused rotate+translate+max into map_pred
  warp_fuse_kernel<<<BSZ * MAPM * MAPM / 256, 256, 0, stream>>>(maps_last, wsf,
                                                                out);
}